// SumLayer_88459146428506
// MI455X (gfx1250) — compile-verified
//
#include <hip/hip_runtime.h>
#include <hip/hip_bf16.h>
#include <stdint.h>

// SumLayer forward: per node, gather 16 rows (1KB each) of element_mars by cids,
// weighted stable log-sum-exp over the 16 children. Memory/gather bound
// (~286 MB moved vs ~0.3 GFLOP), so the win is the CDNA5 DMA gather path:
// TDM tensor_load_to_lds in gather mode (8 indexed rows per instruction),
// double-buffered in LDS across nodes. Fallback: async global->LDS copies.

#define NPB    8      // nodes processed per block (software pipeline depth chunk)
#define NCH    16     // children per node
#define NBATCH 256    // batch columns == blockDim.x

#if defined(__HIP_DEVICE_COMPILE__) && __has_builtin(__builtin_amdgcn_tensor_load_to_lds)
#define USE_TDM 1
#else
#define USE_TDM 0
#endif

typedef unsigned int u32x4 __attribute__((ext_vector_type(4)));
typedef int          i32x8 __attribute__((ext_vector_type(8)));
typedef int          i32x4 __attribute__((ext_vector_type(4)));

#if USE_TDM
// One TDM gather: DMA 8 indexed rows (row = 256 floats) of a row-major
// [n_rows, 256] f32 tensor into LDS at lds_byte, rows packed contiguously.
// D# built per CDNA5 ISA (gather mode, 32-bit indices).
// This toolchain exposes the 6-arg builtin:
//   (u32x4 g0, i32x8 g1, i32x4 g2, i32x4 g3, i32x8 g4_unused, i32 cpol)
__device__ __forceinline__ void tdm_gather8rows(const float* base,
                                                uint32_t lds_byte,
                                                const int* __restrict__ rows,
                                                uint32_t n_rows_total)
{
  const uint64_t ga = (uint64_t)(uintptr_t)base;

  u32x4 g0;
  g0[0] = 0xC0000001u;                                         // count=1 | 32b idx | gather_mode
  g0[1] = lds_byte;                                            // lds_addr (bytes)
  g0[2] = (uint32_t)ga;                                        // global_addr[31:0]
  g0[3] = ((uint32_t)(ga >> 32) & 0x01FFFFFFu) | 0x80000000u;  // global_addr[56:32] | type=2

  i32x8 g1;
  g1[0] = 0x00020000;                                   // data_size=2 (4B); no barrier/iter/pad
  g1[1] = (int)((uint32_t)NBATCH << 16);                // tensor_dim0 = 256  (bits 63:48)
  g1[2] = (int)((n_rows_total & 0xFFFFu) << 16);        // tensor_dim1[15:0] (bits 95:80)
  g1[3] = (int)((n_rows_total >> 16) |
                ((uint32_t)NBATCH << 16));              // tensor_dim1[31:16] | tile_dim0=256
  g1[4] = 8;                                            // tile_dim1 = #gather indices
  g1[5] = NBATCH;                                       // tensor_dim0_stride = 256 elements
  g1[6] = 0;
  g1[7] = 0;

  i32x4 g2, g3;                                         // 32-bit row indices 0..7
  g2[0] = rows[0]; g2[1] = rows[1]; g2[2] = rows[2]; g2[3] = rows[3];
  g3[0] = rows[4]; g3[1] = rows[5]; g3[2] = rows[6]; g3[3] = rows[7];

  const i32x8 g4 = {0, 0, 0, 0, 0, 0, 0, 0};            // unused extra group (NULL per ISA)

  __builtin_amdgcn_tensor_load_to_lds(g0, g1, g2, g3, g4, 0);
}
#endif

// ---- Fallback: CDNA5 async global->LDS copy (ASYNCcnt-tracked). ----
__device__ __forceinline__ void async_ld_b128(uint32_t lds_byte_addr,
                                              uint32_t glb_byte_off,
                                              const float* sbase) {
#if defined(__HIP_DEVICE_COMPILE__)
  asm volatile("global_load_async_to_lds_b128 %0, %1, %2"
               :
               : "v"(lds_byte_addr), "v"(glb_byte_off), "s"(sbase)
               : "memory");
#endif
}

__device__ __forceinline__ void wait_async_0() {
#if defined(__HIP_DEVICE_COMPILE__)
  asm volatile("s_wait_asynccnt 0x0" ::: "memory");
#endif
}

__global__ __launch_bounds__(NBATCH) void sum_layer_fwd(
    const float* __restrict__ element_mars,  // [n_ch_rows, 256]
    const float* __restrict__ params,        // [N_PARAMS]
    const int*   __restrict__ cids,          // [n_nodes, 16]
    const int*   __restrict__ pids,          // [n_nodes, 16]
    const int*   __restrict__ nids,          // [n_nodes]
    float*       __restrict__ out,           // [n_nodes, 256]
    int n_nodes, int n_ch_rows)
{
  // Double-buffered staging tile: 2 x (16 children x 256 floats) = 32 KB LDS.
  __shared__ float buf[2][NCH * NBATCH];

  const int tid  = threadIdx.x;
  const int base = (int)blockIdx.x * NPB;

  const uint32_t lds_base[2] = {
    (uint32_t)(size_t)(&buf[0][0]),   // low 32 bits of LDS-aperture addr = LDS byte offset
    (uint32_t)(size_t)(&buf[1][0])
  };

  // Stage the 16KB gather for node `nn` into buffer `bsel`.
  auto issue = [&](int nn, int bsel) {
    const int* cid_row = cids + (size_t)nn * NCH;
#if USE_TDM
    // TDM ignores EXEC: gate behind a provably-scalar branch so exactly one
    // wave per block issues the DMA (2 instructions cover all 16 rows).
    if (__builtin_amdgcn_readfirstlane(tid) < 32) {
      tdm_gather8rows(element_mars, lds_base[bsel],                     cid_row,
                      (uint32_t)n_ch_rows);
      tdm_gather8rows(element_mars, lds_base[bsel] + 8u * NBATCH * 4u,  cid_row + 8,
                      (uint32_t)n_ch_rows);
    }
#else
    // 4 rounds x 256 lanes x 16B = 16KB; lane covers float4 slot idx:
    // child c = idx>>6, float4-within-row j = idx&63.
    #pragma unroll
    for (int r = 0; r < 4; ++r) {
      const int idx = r * NBATCH + tid;
      const int c   = idx >> 6;
      const int j   = idx & 63;
      const uint32_t goff = (uint32_t)cid_row[c] * (NBATCH * 4u) + (uint32_t)j * 16u;
      const uint32_t loff = lds_base[bsel] + (uint32_t)idx * 16u;
      async_ld_b128(loff, goff, element_mars);
    }
#endif
  };

  auto drain = [&]() {
#if USE_TDM
    __builtin_amdgcn_s_wait_tensorcnt(0);   // trivially true for non-issuing waves
#else
    wait_async_0();
#endif
  };

  if (base < n_nodes) issue(base, 0);

  for (int i = 0; i < NPB; ++i) {
    const int n = base + i;
    if (n >= n_nodes) break;   // block-uniform

    // Drain own DMA/async ops, then block barrier: buffer i&1 is now visible
    // to all waves, and everyone provably finished reading the other buffer.
    drain();
    __syncthreads();

    // Overlap: launch next node's gather into the other buffer.
    if (i + 1 < NPB && n + 1 < n_nodes) issue(n + 1, (i + 1) & 1);

    // Wave-uniform weight gather (scalar s_loads).
    const int* pid_row = pids + (size_t)n * NCH;
    float w[NCH];
    #pragma unroll
    for (int c = 0; c < NCH; ++c) w[c] = params[pid_row[c]];

    // This thread's batch column from the staged tile (bank-conflict free).
    const float* src = &buf[i & 1][0];
    float v[NCH];
    #pragma unroll
    for (int c = 0; c < NCH; ++c) v[c] = src[c * NBATCH + tid];

    // Stable weighted log-sum-exp over children.
    float m = v[0];
    #pragma unroll
    for (int c = 1; c < NCH; ++c) m = fmaxf(m, v[c]);

    float s = 0.0f;
    #pragma unroll
    for (int c = 0; c < NCH; ++c)
      s = __builtin_fmaf(__expf(v[c] - m), w[c], s);

    out[(size_t)nids[n] * NBATCH + tid] = __logf(s) + m;
    // No trailing barrier: next iteration's drain+barrier precedes the issue
    // that overwrites buffer i&1.
  }
}

extern "C" void kernel_launch(void* const* d_in, const int* in_sizes, int n_in,
                              void* d_out, int out_size, void* d_ws, size_t ws_size,
                              hipStream_t stream) {
  // setup_inputs order:
  //   0: node_mars [N,B] (fully overwritten -> unused)
  //   1: element_mars [CH_SIZE, B] f32
  //   2: params [N_PARAMS] f32
  //   3: cids [N, C] i32
  //   4: pids [N, C] i32
  //   5: nids [N] i32
  const float* element_mars = (const float*)d_in[1];
  const float* params       = (const float*)d_in[2];
  const int*   cids         = (const int*)d_in[3];
  const int*   pids         = (const int*)d_in[4];
  const int*   nids         = (const int*)d_in[5];
  float*       out          = (float*)d_out;

  const int n_nodes   = in_sizes[5];
  const int n_ch_rows = in_sizes[1] / NBATCH;
  const int grid      = (n_nodes + NPB - 1) / NPB;

  sum_layer_fwd<<<grid, NBATCH, 0, stream>>>(element_mars, params, cids, pids,
                                             nids, out, n_nodes, n_ch_rows);
}